// Router_3521873183479
// MI455X (gfx1250) — compile-verified
//
#include <hip/hip_runtime.h>
#include <math.h>

// ---------------------------------------------------------------------------
// Switch-router (top-1 MoE gating) for MI455X / gfx1250, wave32.
// Problem constants from the reference: B=4, N=2048, D=4096, E=64.
// ---------------------------------------------------------------------------

typedef float v2f __attribute__((ext_vector_type(2)));
typedef float v8f __attribute__((ext_vector_type(8)));

#define TOKENS   8192      // B*N
#define NBATCH   4
#define NTOK     2048      // tokens per batch
#define DMODEL   4096
#define NEXP     64
#define KCHUNK   128

// Swizzled B tile in LDS: pairs (W[4k4+g][n], W[4k4+g+2][n]) stored
// contiguously so each WMMA B operand is ONE ds_load_b64.
// float offset = k4*SLAB + g*GSTRIDE + 2*n (+slot)
// GSTRIDE = 160 floats == 32 banks (mod 64) -> half-waves hit disjoint banks.
#define GSTRIDE  160
#define SLAB     320       // 2*GSTRIDE
#define BS_FLOATS ((KCHUNK/4) * SLAB)   // 32*320 = 10240 floats = 40 KB

// ---------------------------------------------------------------------------
// Kernel 1: logits[8192][64] = A[8192][4096] @ W[4096][64]
// WMMA f32 16x16x4 (exact f32 math; GEMM is HBM-bound so no precision drop).
// Block = 128 threads = 4 waves -> 128 blocks (2x occupancy spread vs 8-wave
// blocks); wave w owns 16 token rows, all 64 experts (4 accumulators).
// ---------------------------------------------------------------------------
__global__ __launch_bounds__(128)
void k_gemm(const float* __restrict__ A, const float* __restrict__ W,
            float* __restrict__ logits)
{
    __shared__ float Bs[BS_FLOATS];

    const int tid  = threadIdx.x;
    const int wave = tid >> 5;
    const int lane = tid & 31;
    const int g    = lane >> 4;   // half-wave group: selects K pair
    const int nl   = lane & 15;   // N / M position within half-wave
    const int rowBase = blockIdx.x * 64 + wave * 16;
    const int bbase   = g * GSTRIDE + 2 * nl;

    v8f acc0 = {}, acc1 = {}, acc2 = {}, acc3 = {};

    // A layout for V_WMMA_F32_16X16X4_F32: lane l holds row M=l&15,
    // VGPR0/1 = K = 2*(l>>4) and 2*(l>>4)+1.
    const float* aptr = A + (size_t)(rowBase + nl) * DMODEL + 2 * g;

#define WMMA_STEP(av, row) do {                                              \
        const v2f* bp = (const v2f*)&Bs[(row) * SLAB + bbase];               \
        acc0 = __builtin_amdgcn_wmma_f32_16x16x4_f32(false, (av), false,     \
                   bp[0],  (short)0, acc0, false, false);                    \
        acc1 = __builtin_amdgcn_wmma_f32_16x16x4_f32(false, (av), false,     \
                   bp[16], (short)0, acc1, false, false);                    \
        acc2 = __builtin_amdgcn_wmma_f32_16x16x4_f32(false, (av), false,     \
                   bp[32], (short)0, acc2, false, false);                    \
        acc3 = __builtin_amdgcn_wmma_f32_16x16x4_f32(false, (av), false,     \
                   bp[48], (short)0, acc3, false, false);                    \
    } while (0)

    for (int k0 = 0; k0 < DMODEL; k0 += KCHUNK) {
        // Cooperative, coalesced load of W[k0:k0+128, 0:64] into swizzled LDS.
        for (int j = tid; j < KCHUNK * NEXP; j += 128) {
            int k  = j >> 6;
            int n  = j & 63;
            int k4 = k >> 2;
            int gg = k & 1;
            int sl = (k >> 1) & 1;    // which element of the K-pair
            Bs[k4 * SLAB + gg * GSTRIDE + 2 * n + sl] =
                W[(size_t)(k0 + k) * NEXP + n];
        }
        // Prefetch next A chunk while we compute on this one.
        if (k0 + KCHUNK < DMODEL)
            __builtin_prefetch(aptr + k0 + KCHUNK, 0, 0);
        __syncthreads();

        const float* ak = aptr + k0;
        #pragma unroll 2
        for (int k4 = 0; k4 < KCHUNK / 4; k4 += 4) {
            // 16 K-values of A up front -> staggered load waits.
            v2f a0 = *(const v2f*)(ak + 4 * k4);
            v2f a1 = *(const v2f*)(ak + 4 * k4 + 4);
            v2f a2 = *(const v2f*)(ak + 4 * k4 + 8);
            v2f a3 = *(const v2f*)(ak + 4 * k4 + 12);
            WMMA_STEP(a0, k4 + 0);
            WMMA_STEP(a1, k4 + 1);
            WMMA_STEP(a2, k4 + 2);
            WMMA_STEP(a3, k4 + 3);
        }
        __syncthreads();
    }
#undef WMMA_STEP

    // C/D layout: VGPR r, lane l -> M = r + 8*(l>>4), N = l&15.
    float* o = logits + (size_t)rowBase * NEXP + nl;
    #pragma unroll
    for (int r = 0; r < 8; ++r) {
        size_t ro = (size_t)(r + 8 * g) * NEXP;
        o[ro]      = acc0[r];
        o[ro + 16] = acc1[r];
        o[ro + 32] = acc2[r];
        o[ro + 48] = acc3[r];
    }
}

// ---------------------------------------------------------------------------
// Kernel 2: per-token softmax stats: argmax index, gate = max prob, logsumexp.
// ---------------------------------------------------------------------------
__global__ __launch_bounds__(256)
void k_softmax(const float* __restrict__ logits, float* __restrict__ lse,
               float* __restrict__ gate, int* __restrict__ eidx)
{
    int t = blockIdx.x * 256 + threadIdx.x;
    if (t >= TOKENS) return;
    const float4* l4 = (const float4*)(logits + (size_t)t * NEXP);

    float mx = -3.4e38f;
    int   mi = 0;
    #pragma unroll
    for (int j = 0; j < 16; ++j) {
        float4 v = l4[j];
        // first-max tie-breaking to match jnp.argmax
        if (v.x > mx) { mx = v.x; mi = 4 * j + 0; }
        if (v.y > mx) { mx = v.y; mi = 4 * j + 1; }
        if (v.z > mx) { mx = v.z; mi = 4 * j + 2; }
        if (v.w > mx) { mx = v.w; mi = 4 * j + 3; }
    }
    float s = 0.f;
    #pragma unroll
    for (int j = 0; j < 16; ++j) {
        float4 v = l4[j];
        s += expf(v.x - mx) + expf(v.y - mx) + expf(v.z - mx) + expf(v.w - mx);
    }
    lse[t]  = mx + logf(s);
    gate[t] = 1.0f / s;          // exp(mx - lse)
    eidx[t] = mi;
}

// ---------------------------------------------------------------------------
// Kernel 3: per-(batch, expert) ordered scan over tokens.
//   pos[token]  = 0-based slot index within its argmax expert (token order)
//   cnt[b*64+e] = #tokens dispatched to (b,e)      (density numerator)
//   psum[b*64+e]= sum_n softmax_prob[b,n,e]        (density_proxy numerator)
// One block per (b,e): 256 blocks, 256 threads, 8 ordered chunks of 256.
// ---------------------------------------------------------------------------
__global__ __launch_bounds__(256)
void k_scan(const int* __restrict__ eidx, const float* __restrict__ logits,
            const float* __restrict__ lse, int* __restrict__ pos,
            float* __restrict__ cnt, float* __restrict__ psum)
{
    const int b = blockIdx.x >> 6;
    const int e = blockIdx.x & 63;
    const int t = threadIdx.x;
    __shared__ int   s[256];
    __shared__ float sf[256];

    int   carry = 0;
    float pacc  = 0.f;
    const int base = b * NTOK;

    for (int chunk = 0; chunk < NTOK; chunk += 256) {
        const int tok = base + chunk + t;
        const int m   = (eidx[tok] == e) ? 1 : 0;
        pacc += expf(logits[(size_t)tok * NEXP + e] - lse[tok]);

        s[t] = m;
        __syncthreads();
        // Hillis-Steele inclusive scan over 256 entries.
        #pragma unroll
        for (int off = 1; off < 256; off <<= 1) {
            int v = (t >= off) ? s[t - off] : 0;
            __syncthreads();
            s[t] += v;
            __syncthreads();
        }
        if (m) pos[tok] = carry + s[t] - 1;   // each token written exactly once
        carry += s[255];
        __syncthreads();
    }

    sf[t] = pacc;
    __syncthreads();
    #pragma unroll
    for (int off = 128; off > 0; off >>= 1) {
        if (t < off) sf[t] += sf[t + off];
        __syncthreads();
    }
    if (t == 0) {
        cnt[blockIdx.x]  = (float)carry;
        psum[blockIdx.x] = sf[0];
    }
}

// ---------------------------------------------------------------------------
// Kernel 4: write dispatch [b,n,e,c] and combine tensors.
// One block per token; fully coalesced float4 stores (c divisible by 4).
// Dominant memory cost of the whole problem -> pure streaming stores.
// ---------------------------------------------------------------------------
__global__ __launch_bounds__(128)
void k_outputs(const int* __restrict__ eidx, const int* __restrict__ pos,
               const float* __restrict__ gate, float4* __restrict__ out,
               int cap, size_t combOfs4)
{
    const int token = blockIdx.x;
    const int t     = threadIdx.x;
    const int ei    = eidx[token];
    const int p     = pos[token];
    const float gv  = gate[token];
    const int cap4  = cap >> 2;          // float4 per (e) row
    const int per   = NEXP * cap4;       // float4 per token
    const size_t basev = (size_t)token * per;
    const bool ok = (p < cap);

    for (int f = t; f < per; f += 128) {
        const int e  = f / cap4;
        const int c0 = (f - e * cap4) * 4;
        float4 d = make_float4(0.f, 0.f, 0.f, 0.f);
        if (e == ei && ok && p >= c0 && p < c0 + 4)
            ((float*)&d)[p - c0] = 1.0f;
        out[basev + f] = d;
        float4 c = make_float4(d.x * gv, d.y * gv, d.z * gv, d.w * gv);
        out[combOfs4 + basev + f] = c;
    }
}

// ---------------------------------------------------------------------------
// Kernel 5: scalars. aux = mean(density*proxy)*E^2 ; z = mean(lse^2).
// ---------------------------------------------------------------------------
__global__ __launch_bounds__(256)
void k_losses(const float* __restrict__ lse, const float* __restrict__ cnt,
              const float* __restrict__ psum, float* __restrict__ out,
              size_t auxIdx)
{
    __shared__ float sf[256];
    const int t = threadIdx.x;

    float z = 0.f;
    for (int tok = t; tok < TOKENS; tok += 256) {
        float v = lse[tok];
        z += v * v;
    }
    sf[t] = z;
    __syncthreads();
    #pragma unroll
    for (int off = 128; off > 0; off >>= 1) {
        if (t < off) sf[t] += sf[t + off];
        __syncthreads();
    }
    const float ztot = sf[0];
    __syncthreads();

    // 256 (b,e) pairs, one per thread.
    const float inv = 1.0f / (float)NTOK;
    sf[t] = (cnt[t] * inv) * (psum[t] * inv);
    __syncthreads();
    #pragma unroll
    for (int off = 128; off > 0; off >>= 1) {
        if (t < off) sf[t] += sf[t + off];
        __syncthreads();
    }
    if (t == 0) {
        // mean over 256 pairs * E^2 = (sum/256)*4096 = sum*16
        out[auxIdx]     = sf[0] * 16.0f;
        out[auxIdx + 1] = ztot * (1.0f / (float)TOKENS);
    }
}

// ---------------------------------------------------------------------------
// Host launcher.
// ---------------------------------------------------------------------------
extern "C" void kernel_launch(void* const* d_in, const int* in_sizes, int n_in,
                              void* d_out, int out_size, void* d_ws, size_t ws_size,
                              hipStream_t stream)
{
    const float* A = (const float*)d_in[0];   // [4,2048,4096]
    const float* W = (const float*)d_in[1];   // [4096,64]
    // expert_capacity recovered on host from out_size (device scalar is
    // unreadable under graph capture): out = 2*B*N*E*cap + 2
    const int cap = (out_size - 2) / (2 * TOKENS * NEXP);

    // Workspace layout (bytes):
    char* ws = (char*)d_ws;
    float* logits = (float*)ws;                                  // 8192*64
    float* lse    = (float*)(ws + (size_t)TOKENS * NEXP * 4);    // 8192
    float* gate   = lse  + TOKENS;                               // 8192
    int*   eidx   = (int*)(gate + TOKENS);                       // 8192
    int*   pos    = eidx + TOKENS;                               // 8192
    float* cnt    = (float*)(pos + TOKENS);                      // 256
    float* psum   = cnt + NBATCH * NEXP;                         // 256

    k_gemm   <<<TOKENS / 64, 128, 0, stream>>>(A, W, logits);
    k_softmax<<<TOKENS / 256, 256, 0, stream>>>(logits, lse, gate, eidx);
    k_scan   <<<NBATCH * NEXP, 256, 0, stream>>>(eidx, logits, lse, pos, cnt, psum);

    const size_t nPer  = (size_t)TOKENS * NEXP * cap;   // dispatch element count
    k_outputs<<<TOKENS, 128, 0, stream>>>(eidx, pos, gate, (float4*)d_out,
                                          cap, nPer / 4);
    k_losses <<<1, 256, 0, stream>>>(lse, cnt, psum, (float*)d_out, 2 * nPer);
}